// graph2linegraph_12463995093127
// MI455X (gfx1250) — compile-verified
//
#include <hip/hip_runtime.h>
#include <hip/hip_bf16.h>
#include <stdint.h>

#define N_NODES  4096
#define N_EDGES  16384
#define DFEAT    128
#define NCHUNK   8             // j-dimension split for the O(E^2) mask step
#define CHUNK    2048          // edges staged in LDS per block (2 x 8KB)
#define IDX_BASE 4194304       // E * 2D floats of x_new precede the index block

typedef float v4f __attribute__((ext_vector_type(4)));

// ---- CDNA5 async global->LDS staging (ASYNCcnt path) -----------------------
__device__ __forceinline__ unsigned lds_off_u32(const void* p) {
  // generic -> LDS addrspace(3) -> 32-bit byte offset
  return (unsigned)(unsigned long long)(__attribute__((address_space(3))) const char*)p;
}

__device__ __forceinline__ void async_copy_b128(unsigned lds, const void* g) {
  asm volatile("global_load_async_to_lds_b128 %0, %1, off"
               :: "v"(lds), "v"((unsigned long long)(uintptr_t)g)
               : "memory");
}

__device__ __forceinline__ void wait_async0() {
  asm volatile("s_wait_asynccnt 0" ::: "memory");
}

// ---- Kernel 1: per-(row, j-chunk) match counts -----------------------------
// grid = (E/256, NCHUNK); cnt is [E][NCHUNK] row-major so a flat exclusive
// scan over it yields offsets in exact jnp.nonzero lexicographic order.
__global__ void lg_count_kernel(const int* __restrict__ ei, int* __restrict__ cnt) {
  __shared__ int s_src[CHUNK];
  __shared__ int s_dst[CHUNK];
  const int tid  = threadIdx.x;
  const int i    = blockIdx.x * 256 + tid;
  const int base = blockIdx.y * CHUNK;
  const int* src = ei;
  const int* dst = ei + N_EDGES;
  const int my_src = src[i];
  const int my_dst = dst[i];
#pragma unroll
  for (int r = 0; r < CHUNK / (256 * 4); ++r) {
    const int idx = (r * 256 + tid) * 4;
    async_copy_b128(lds_off_u32(&s_src[idx]), src + base + idx);
    async_copy_b128(lds_off_u32(&s_dst[idx]), dst + base + idx);
  }
  wait_async0();            // each wave drains its own ASYNCcnt
  __syncthreads();          // then LDS is visible workgroup-wide
  int c = 0;
  for (int j4 = 0; j4 < CHUNK; j4 += 4) {
    const int4 sj = *reinterpret_cast<const int4*>(&s_src[j4]);  // ds_load_b128, broadcast
    const int4 dj = *reinterpret_cast<const int4*>(&s_dst[j4]);
    c += (sj.x == my_dst) & (dj.x != my_src);
    c += (sj.y == my_dst) & (dj.y != my_src);
    c += (sj.z == my_dst) & (dj.z != my_src);
    c += (sj.w == my_dst) & (dj.w != my_src);
  }
  cnt[i * NCHUNK + blockIdx.y] = c;
}

// ---- Kernel 2: exclusive scan over E*NCHUNK = 131072 counts (1 block) ------
__global__ void lg_scan_kernel(int* __restrict__ off) {
  __shared__ int part[1024];
  const int t   = threadIdx.x;
  const int PER = (N_EDGES * NCHUNK) / 1024;  // 128 per thread
  int s = 0;
  for (int k = 0; k < PER; ++k) s += off[t * PER + k];
  part[t] = s;
  __syncthreads();
  for (int d = 1; d < 1024; d <<= 1) {
    const int add = (t >= d) ? part[t - d] : 0;
    __syncthreads();
    part[t] += add;
    __syncthreads();
  }
  int run = (t == 0) ? 0 : part[t - 1];
  for (int k = 0; k < PER; ++k) {
    const int c = off[t * PER + k];
    off[t * PER + k] = run;
    run += c;
  }
}

// ---- Kernel 3: emit (i,j) pairs in nonzero (lexicographic) order -----------
__global__ void lg_emit_kernel(const int* __restrict__ ei, const int* __restrict__ off,
                               int* __restrict__ pairs, float* __restrict__ out, int M) {
  __shared__ int s_src[CHUNK];
  __shared__ int s_dst[CHUNK];
  const int tid  = threadIdx.x;
  const int i    = blockIdx.x * 256 + tid;
  const int base = blockIdx.y * CHUNK;
  const int* src = ei;
  const int* dst = ei + N_EDGES;
  const int my_src = src[i];
  const int my_dst = dst[i];
#pragma unroll
  for (int r = 0; r < CHUNK / (256 * 4); ++r) {
    const int idx = (r * 256 + tid) * 4;
    async_copy_b128(lds_off_u32(&s_src[idx]), src + base + idx);
    async_copy_b128(lds_off_u32(&s_dst[idx]), dst + base + idx);
  }
  wait_async0();
  __syncthreads();
  int p = off[i * NCHUNK + blockIdx.y];
  for (int j4 = 0; j4 < CHUNK; j4 += 4) {
    const int4 sj = *reinterpret_cast<const int4*>(&s_src[j4]);
    const int4 dj = *reinterpret_cast<const int4*>(&s_dst[j4]);
    const int jb = base + j4;
#pragma unroll
    for (int q = 0; q < 4; ++q) {
      const int sv = (q == 0) ? sj.x : (q == 1) ? sj.y : (q == 2) ? sj.z : sj.w;
      const int dv = (q == 0) ? dj.x : (q == 1) ? dj.y : (q == 2) ? dj.z : dj.w;
      if ((sv == my_dst) && (dv != my_src)) {
        const int j = jb + q;
        __builtin_nontemporal_store((float)i, out + IDX_BASE + p);
        __builtin_nontemporal_store((float)j, out + IDX_BASE + M + p);
        pairs[2 * p]     = i;
        pairs[2 * p + 1] = j;
        ++p;
      }
    }
  }
}

// ---- Kernel 4: x_new = 0.5*([x[src]|x[dst]] + [ea|ea]) ; wave32 per edge ---
__global__ void lg_xnew_kernel(const float* __restrict__ x, const int* __restrict__ ei,
                               const float* __restrict__ ea, float* __restrict__ out) {
  const int tid  = threadIdx.x;
  const int lane = tid & 31;
  const int e    = blockIdx.x * 8 + (tid >> 5);
  const int s = ei[e];
  const int t = ei[N_EDGES + e];
  const int d = lane * 4;
  __builtin_prefetch(x + (size_t)s * DFEAT, 0, 3);   // global_prefetch_b8
  __builtin_prefetch(x + (size_t)t * DFEAT, 0, 3);
  const v4f xs = *reinterpret_cast<const v4f*>(x + (size_t)s * DFEAT + d);
  const v4f xt = *reinterpret_cast<const v4f*>(x + (size_t)t * DFEAT + d);
  const v4f a  = *reinterpret_cast<const v4f*>(ea + (size_t)e * DFEAT + d);
  const v4f o0 = 0.5f * (xs + a);
  const v4f o1 = 0.5f * (xt + a);
  __builtin_nontemporal_store(o0, (v4f*)(out + (size_t)e * 256 + d));        // streaming
  __builtin_nontemporal_store(o1, (v4f*)(out + (size_t)e * 256 + 128 + d));
}

// ---- Kernel 5: edge_attr_new per line-graph edge ; wave32 per pair ---------
__global__ void lg_eattr_kernel(const float* __restrict__ x, const int* __restrict__ ei,
                                const float* __restrict__ ea, const int* __restrict__ pairs,
                                float* __restrict__ out, int M) {
  const int tid  = threadIdx.x;
  const int lane = tid & 31;
  const int p    = blockIdx.x * 8 + (tid >> 5);
  if (p >= M) return;
  const int i = pairs[2 * p];
  const int j = pairs[2 * p + 1];
  const int shared_n = ei[N_EDGES + i];               // dst[i]
  const int d = lane * 4;
  __builtin_prefetch(x + (size_t)shared_n * DFEAT, 0, 3);
  const v4f xs = *reinterpret_cast<const v4f*>(x + (size_t)shared_n * DFEAT + d);
  const v4f ai = *reinterpret_cast<const v4f*>(ea + (size_t)i * DFEAT + d);
  const v4f aj = *reinterpret_cast<const v4f*>(ea + (size_t)j * DFEAT + d);
  const size_t baseo = (size_t)IDX_BASE + 2ull * (unsigned)M + (size_t)p * 256 + d;
  __builtin_nontemporal_store(0.5f * (xs + ai), (v4f*)(out + baseo));
  __builtin_nontemporal_store(0.5f * (xs + aj), (v4f*)(out + baseo + 128));
}

// ---- Host launch -----------------------------------------------------------
extern "C" void kernel_launch(void* const* d_in, const int* in_sizes, int n_in,
                              void* d_out, int out_size, void* d_ws, size_t ws_size,
                              hipStream_t stream) {
  (void)in_sizes; (void)n_in; (void)ws_size;
  const float* x  = (const float*)d_in[0];
  const int*   ei = (const int*)d_in[1];      // [2, E] row-major int32
  const float* ea = (const float*)d_in[2];
  float* out = (float*)d_out;

  int* off   = (int*)d_ws;                    // E*NCHUNK counts -> exclusive offsets
  int* pairs = off + (N_EDGES * NCHUNK + 64); // 2*M entries

  // out_size = E*2D + M*(2 + 2D)  =>  M known on host
  const long long Mll = ((long long)out_size - (long long)IDX_BASE) / 258LL;
  const int M = (int)(Mll > 0 ? Mll : 0);

  lg_count_kernel<<<dim3(N_EDGES / 256, NCHUNK), dim3(256), 0, stream>>>(ei, off);
  lg_scan_kernel <<<dim3(1),                     dim3(1024), 0, stream>>>(off);
  lg_emit_kernel <<<dim3(N_EDGES / 256, NCHUNK), dim3(256), 0, stream>>>(ei, off, pairs, out, M);
  lg_xnew_kernel <<<dim3(N_EDGES / 8),           dim3(256), 0, stream>>>(x, ei, ea, out);
  if (M > 0)
    lg_eattr_kernel<<<dim3((M + 7) / 8),         dim3(256), 0, stream>>>(x, ei, ea, pairs, out, M);
}